// NSF_Mdiff_CNNcond_84035330113542
// MI455X (gfx1250) — compile-verified
//
#include <hip/hip_runtime.h>
#include <math.h>

// Problem constants (match reference)
#define DIMV   8
#define NCONDV 16
#define KB     5
#define NF     2
#define HIDV   8
#define CMAXV  23
#define TBF    3.0f
#define MINWF  0.001f
#define MINDF  0.001f
#define HSTR   17   // LDS row stride (odd -> conflict free across 64 banks)

typedef __attribute__((ext_vector_type(2))) float v2f;
typedef __attribute__((ext_vector_type(8))) float v8f;

// ---------------------------------------------------------------------------
// Wave-local LDS fence: per the CDNA5 ISA, LDS ops from the same wave execute
// in order at the LDS unit, so cross-lane store->load within one wave's
// private buffer needs only (a) program-order issue and (b) a compiler fence
// so clang cannot reorder ds ops it cannot prove non-aliasing for.
// ---------------------------------------------------------------------------
__device__ __forceinline__ void lds_fence() {
  __asm__ __volatile__("s_wait_dscnt 0x0" ::: "memory");
}

// WMMA: D = A(16x4 f32) * B(4x16 f32) + C(16x16 f32)
__device__ __forceinline__ v8f wmma4(v2f a, v2f b, v8f c) {
  return __builtin_amdgcn_wmma_f32_16x16x4_f32(false, a, false, b, (short)0, c,
                                               false, false);
}

__device__ __forceinline__ v8f splat8(float v) {
  v8f c;
  c[0] = v; c[1] = v; c[2] = v; c[3] = v;
  c[4] = v; c[5] = v; c[6] = v; c[7] = v;
  return c;
}

// tanh(x) = 1 - 2/(exp(2x)+1): v_exp + v_rcp, saturates correctly at +-1.
__device__ __forceinline__ float fast_tanh(float x) {
  float e = __expf(2.0f * x);
  return 1.0f - 2.0f * __builtin_amdgcn_rcpf(e + 1.0f);
}

// B fragment for one K-chunk of W^T (W is [nrows x ncols] row-major):
// B[k][n] = W[n][k]; VGPR j holds K = kbase + j + 2*(lane/16), N = lane%16.
__device__ __forceinline__ v2f load_bfrag(const float* __restrict__ W,
                                          int nrows, int ncols, int kbase,
                                          int lane) {
  int n = lane & 15;
  int k = kbase + ((lane >> 4) << 1);
  v2f b;
  b.x = (n < nrows && k < ncols) ? W[n * ncols + k] : 0.0f;
  b.y = (n < nrows && (k + 1) < ncols) ? W[n * ncols + k + 1] : 0.0f;
  return b;
}

// ---------------------------------------------------------------------------
// Packed evaluation of BOTH flow nets of one dim over a 32-row tile.
// Columns 0-7 of the 16-wide WMMA carry flow0, columns 8-15 carry flow1
// (block-diagonal B in layer 2), so every output column is useful.
// ---------------------------------------------------------------------------
__device__ __forceinline__ void eval_flow_pair(
    const v2f aM[2][6],
    const float* __restrict__ W1a, const float* __restrict__ b1a,
    const float* __restrict__ W1b, const float* __restrict__ b1b,
    const float* __restrict__ W2a, const float* __restrict__ b2a,
    const float* __restrict__ W2b, const float* __restrict__ b2b,
    const float* __restrict__ W3a, const float* __restrict__ b3a,
    const float* __restrict__ W3b, const float* __restrict__ b3b,
    float* __restrict__ hbuf, int lane, v8f cout0[2], v8f cout1[2]) {
  const int n = lane & 15;
  const int nn = n & 7;
  const int kh = (lane >> 4) << 1;
  const int mrow = (lane >> 4) << 3;
  const bool lo = (n < 8);

  // ---- layer 1: both flows side by side in N ----
  const float* W1s = lo ? W1a : W1b;
  const float* b1s = lo ? b1a : b1b;
  v2f B1[6];
#pragma unroll
  for (int c = 0; c < 6; ++c) {
    int k = 4 * c + kh;
    B1[c].x = (k < CMAXV) ? W1s[nn * CMAXV + k] : 0.0f;
    B1[c].y = (k + 1 < CMAXV) ? W1s[nn * CMAXV + k + 1] : 0.0f;
  }
  float bv1 = b1s[nn];
  v8f h1[2];
#pragma unroll
  for (int t = 0; t < 2; ++t) {
    v8f acc = splat8(bv1);
#pragma unroll
    for (int c = 0; c < 6; ++c) acc = wmma4(aM[t][c], B1[c], acc);
#pragma unroll
    for (int v = 0; v < 8; ++v) acc[v] = fast_tanh(acc[v]);
    h1[t] = acc;
  }
  lds_fence();
#pragma unroll
  for (int t = 0; t < 2; ++t)
#pragma unroll
    for (int v = 0; v < 8; ++v)
      hbuf[(t * 16 + v + mrow) * HSTR + n] = h1[t][v];
  lds_fence();

  // ---- layer 2: K = 16 (cols 0-7 flow0, 8-15 flow1), block-diagonal B ----
  v2f A2[2][4];
#pragma unroll
  for (int t = 0; t < 2; ++t)
#pragma unroll
    for (int k2 = 0; k2 < 4; ++k2) {
      int row = t * 16 + (lane & 15);
      int col = 4 * k2 + kh;
      A2[t][k2].x = hbuf[row * HSTR + col];
      A2[t][k2].y = hbuf[row * HSTR + col + 1];
    }
  const float* W2s = lo ? W2a : W2b;
  const float* b2s = lo ? b2a : b2b;
  v2f B2[4];
#pragma unroll
  for (int k2 = 0; k2 < 4; ++k2) {
    int k0 = 4 * k2 + kh;
    int k1 = k0 + 1;
    B2[k2].x = ((k0 >> 3) == (n >> 3)) ? W2s[nn * HIDV + (k0 & 7)] : 0.0f;
    B2[k2].y = ((k1 >> 3) == (n >> 3)) ? W2s[nn * HIDV + (k1 & 7)] : 0.0f;
  }
  float bv2 = b2s[nn];
  v8f h2[2];
#pragma unroll
  for (int t = 0; t < 2; ++t) {
    v8f acc = splat8(bv2);
#pragma unroll
    for (int k2 = 0; k2 < 4; ++k2) acc = wmma4(A2[t][k2], B2[k2], acc);
#pragma unroll
    for (int v = 0; v < 8; ++v) acc[v] = fast_tanh(acc[v]);
    h2[t] = acc;
  }
  lds_fence();
#pragma unroll
  for (int t = 0; t < 2; ++t)
#pragma unroll
    for (int v = 0; v < 8; ++v)
      hbuf[(t * 16 + v + mrow) * HSTR + n] = h2[t][v];
  lds_fence();

  // ---- layer 3: per-flow (14 outputs each do not pack into 16 cols) ----
#pragma unroll
  for (int f = 0; f < 2; ++f) {
    const float* W3f = (f == 0) ? W3a : W3b;
    const float* b3f = (f == 0) ? b3a : b3b;
    v2f A3[2][2];
#pragma unroll
    for (int t = 0; t < 2; ++t)
#pragma unroll
      for (int k = 0; k < 2; ++k) {
        int row = t * 16 + (lane & 15);
        int col = 8 * f + 4 * k + kh;
        A3[t][k].x = hbuf[row * HSTR + col];
        A3[t][k].y = hbuf[row * HSTR + col + 1];
      }
    v2f B3[2];
#pragma unroll
    for (int k = 0; k < 2; ++k)
      B3[k] = load_bfrag(W3f, 3 * KB - 1, HIDV, 4 * k, lane);
    float bv3 = (n < 3 * KB - 1) ? b3f[n] : 0.0f;
#pragma unroll
    for (int t = 0; t < 2; ++t) {
      v8f acc = splat8(bv3);
      acc = wmma4(A3[t][0], B3[0], acc);
      acc = wmma4(A3[t][1], B3[1], acc);
      if (f == 0) cout0[t] = acc; else cout1[t] = acc;
    }
  }
}

// ---------------------------------------------------------------------------
// Gaussianizer net (23 -> 8 -> 8 -> 2), single net over the 32-row tile.
// ---------------------------------------------------------------------------
__device__ __forceinline__ void eval_gnet(const v2f aM[2][6],
                                          const float* __restrict__ W1,
                                          const float* __restrict__ b1,
                                          const float* __restrict__ W2,
                                          const float* __restrict__ b2,
                                          const float* __restrict__ W3,
                                          const float* __restrict__ b3,
                                          float* __restrict__ hbuf, int lane,
                                          v8f cout[2]) {
  const int n = lane & 15;
  const int kh = (lane >> 4) << 1;
  const int mrow = (lane >> 4) << 3;

  v2f B1[6];
#pragma unroll
  for (int c = 0; c < 6; ++c) B1[c] = load_bfrag(W1, HIDV, CMAXV, 4 * c, lane);
  float bv1 = (n < HIDV) ? b1[n] : 0.0f;
  v8f h1[2];
#pragma unroll
  for (int t = 0; t < 2; ++t) {
    v8f acc = splat8(bv1);
#pragma unroll
    for (int c = 0; c < 6; ++c) acc = wmma4(aM[t][c], B1[c], acc);
#pragma unroll
    for (int v = 0; v < 8; ++v) acc[v] = fast_tanh(acc[v]);
    h1[t] = acc;
  }
  lds_fence();
#pragma unroll
  for (int t = 0; t < 2; ++t)
#pragma unroll
    for (int v = 0; v < 8; ++v)
      hbuf[(t * 16 + v + mrow) * HSTR + n] = h1[t][v];
  lds_fence();

  v2f A2[2][2];
#pragma unroll
  for (int t = 0; t < 2; ++t)
#pragma unroll
    for (int k = 0; k < 2; ++k) {
      int row = t * 16 + (lane & 15);
      int col = 4 * k + kh;
      A2[t][k].x = hbuf[row * HSTR + col];
      A2[t][k].y = hbuf[row * HSTR + col + 1];
    }
  v2f B2[2];
#pragma unroll
  for (int k = 0; k < 2; ++k) B2[k] = load_bfrag(W2, HIDV, HIDV, 4 * k, lane);
  float bv2 = (n < HIDV) ? b2[n] : 0.0f;
  v8f h2[2];
#pragma unroll
  for (int t = 0; t < 2; ++t) {
    v8f acc = splat8(bv2);
    acc = wmma4(A2[t][0], B2[0], acc);
    acc = wmma4(A2[t][1], B2[1], acc);
#pragma unroll
    for (int v = 0; v < 8; ++v) acc[v] = fast_tanh(acc[v]);
    h2[t] = acc;
  }
  lds_fence();
#pragma unroll
  for (int t = 0; t < 2; ++t)
#pragma unroll
    for (int v = 0; v < 8; ++v)
      hbuf[(t * 16 + v + mrow) * HSTR + n] = h2[t][v];
  lds_fence();

  v2f A3[2][2];
#pragma unroll
  for (int t = 0; t < 2; ++t)
#pragma unroll
    for (int k = 0; k < 2; ++k) {
      int row = t * 16 + (lane & 15);
      int col = 4 * k + kh;
      A3[t][k].x = hbuf[row * HSTR + col];
      A3[t][k].y = hbuf[row * HSTR + col + 1];
    }
  v2f B3[2];
#pragma unroll
  for (int k = 0; k < 2; ++k) B3[k] = load_bfrag(W3, 2, HIDV, 4 * k, lane);
  float bv3 = (n < 2) ? b3[n] : 0.0f;
#pragma unroll
  for (int t = 0; t < 2; ++t) {
    v8f acc = splat8(bv3);
    acc = wmma4(A3[t][0], B3[0], acc);
    acc = wmma4(A3[t][1], B3[1], acc);
    cout[t] = acc;
  }
}

// ---------------------------------------------------------------------------
// Scalar spline (per lane == per sample), replicating the reference exactly
// (including the double softmax / double softplus).
// ---------------------------------------------------------------------------
__device__ __forceinline__ void softmax5(const float* v, float* o) {
  float m = v[0];
#pragma unroll
  for (int i = 1; i < 5; ++i) m = fmaxf(m, v[i]);
  float e[5], s = 0.0f;
#pragma unroll
  for (int i = 0; i < 5; ++i) {
    e[i] = __expf(v[i] - m);
    s += e[i];
  }
  float inv = __builtin_amdgcn_rcpf(s);
#pragma unroll
  for (int i = 0; i < 5; ++i) o[i] = e[i] * inv;
}

__device__ __forceinline__ float softplusf(float x) {
  return fmaxf(x, 0.0f) + log1pf(__expf(-fabsf(x)));
}

__device__ __forceinline__ float sel6(const float* a, int idx) {
  float r = a[0];
#pragma unroll
  for (int k = 1; k < 6; ++k) r = (idx == k) ? a[k] : r;
  return r;
}

__device__ __forceinline__ float rqs_step(float x, const float* o,
                                          float* ld_sum) {
  const float twoTB = 2.0f * TBF;
  float uw[5], uh[5];
  softmax5(o, uw);
  softmax5(o + 5, uh);
#pragma unroll
  for (int i = 0; i < 5; ++i) {
    uw[i] *= twoTB;
    uh[i] *= twoTB;
  }
  float wsm[5], hsm[5];  // reference applies softmax again inside _rqs_fwd
  softmax5(uw, wsm);
  softmax5(uh, hsm);

  float cw[6], ch[6];
  cw[0] = -TBF;
  ch[0] = -TBF;
  float accw = 0.0f, acch = 0.0f;
#pragma unroll
  for (int i = 0; i < 5; ++i) {
    accw += MINWF + (1.0f - 5.0f * MINWF) * wsm[i];
    acch += MINWF + (1.0f - 5.0f * MINWF) * hsm[i];
    cw[i + 1] = twoTB * accw - TBF;
    ch[i + 1] = twoTB * acch - TBF;
  }
  cw[5] = TBF;
  ch[5] = TBF;

  // boundary derivs are exactly 1.0 (MIND + softplus(pad const)); middle
  // values get softplus twice (caller + _rqs_fwd), as in the reference
  float dv[6];
  dv[0] = 1.0f;
  dv[5] = 1.0f;
#pragma unroll
  for (int i = 0; i < 4; ++i)
    dv[i + 1] = MINDF + softplusf(softplusf(o[10 + i]));

  float xc = fminf(fmaxf(x, -TBF), TBF);
  int idx = 0;
#pragma unroll
  for (int i = 1; i <= 4; ++i) idx += (xc >= cw[i]) ? 1 : 0;  // idx in [0,4]

  float icw = sel6(cw, idx);
  float ibw = sel6(cw, idx + 1) - icw;
  float ich = sel6(ch, idx);
  float ih = sel6(ch, idx + 1) - ich;
  float idelta = ih / ibw;
  float d0 = sel6(dv, idx);
  float d1 = sel6(dv, idx + 1);

  float th = (xc - icw) / ibw;
  float t2 = th * th;
  float omth = 1.0f - th;
  float tomt = th * omth;
  float den = idelta + (d0 + d1 - 2.0f * idelta) * tomt;
  float z = ich + ih * (idelta * t2 + d0 * tomt) / den;
  float num = d1 * t2 + 2.0f * idelta * tomt + d0 * omth * omth;
  float ld = __logf(idelta * idelta * num) - 2.0f * __logf(den);

  bool inside = (x >= -TBF) && (x <= TBF);
  *ld_sum += inside ? ld : 0.0f;
  return inside ? z : x;
}

// ---------------------------------------------------------------------------
// Main kernel: one wave per 32-sample tile; waves fully independent.
// ---------------------------------------------------------------------------
__global__ __launch_bounds__(256) void nsf_logp_kernel(
    const float* __restrict__ x_inp, const float* __restrict__ cond_inp,
    const float* __restrict__ Wi1, const float* __restrict__ bi1,
    const float* __restrict__ Wi2, const float* __restrict__ bi2,
    const float* __restrict__ Wi3, const float* __restrict__ bi3,
    const float* __restrict__ Wf1, const float* __restrict__ bf1,
    const float* __restrict__ Wf2, const float* __restrict__ bf2,
    const float* __restrict__ Wf3, const float* __restrict__ bf3,
    float* __restrict__ out, int N) {
  __shared__ float s_h[8][32 * HSTR];  // per-wave staging tile
  __shared__ float s_mu[8][32];
  __shared__ float s_ls[8][32];

  const int lane = threadIdx.x & 31;
  const int wv = threadIdx.x >> 5;
  const int tileBase = blockIdx.x * 256 + wv * 32;
  const int sample = tileBase + lane;
  const int sclamp = (sample < N) ? sample : (N - 1);

  float* hbuf = s_h[wv];

  const int r16 = lane & 15;
  const int kh = (lane >> 4) << 1;
  const int mrow = (lane >> 4) << 3;

  // Unmasked A fragments for the 24-col [cond(16) | x(8)] matrix.
  v2f aU[2][6];
#pragma unroll
  for (int t = 0; t < 2; ++t) {
    int row = tileBase + t * 16 + r16;
    if (row >= N) row = N - 1;
#pragma unroll
    for (int c = 0; c < 4; ++c) {
      int col = 4 * c + kh;
      aU[t][c] = *(const v2f*)(cond_inp + (size_t)row * NCONDV + col);
    }
#pragma unroll
    for (int c = 4; c < 6; ++c) {
      int col = 4 * c + kh - 16;
      aU[t][c] = *(const v2f*)(x_inp + (size_t)row * DIMV + col);
    }
  }

  float total = 0.0f;

  for (int jd = 0; jd < DIMV; ++jd) {
    // Mask autoregressive x columns >= jd (cols beyond 16+jd become zero).
    v2f aM[2][6];
    const int limit = NCONDV + jd;
#pragma unroll
    for (int t = 0; t < 2; ++t) {
#pragma unroll
      for (int c = 0; c < 4; ++c) aM[t][c] = aU[t][c];
#pragma unroll
      for (int c = 4; c < 6; ++c) {
        int col = 4 * c + kh;
        v2f a = aU[t][c];
        a.x = (col < limit) ? a.x : 0.0f;
        a.y = (col + 1 < limit) ? a.y : 0.0f;
        aM[t][c] = a;
      }
    }

    float x = x_inp[(size_t)sclamp * DIMV + jd];
    float ld_sum = 0.0f;

    // ---- both flow nets in one packed pass ----
    const int fi0 = jd * NF;
    const int fi1 = fi0 + 1;
    v8f c0[2], c1[2];
    eval_flow_pair(
        aM, Wf1 + (size_t)fi0 * HIDV * CMAXV, bf1 + (size_t)fi0 * HIDV,
        Wf1 + (size_t)fi1 * HIDV * CMAXV, bf1 + (size_t)fi1 * HIDV,
        Wf2 + (size_t)fi0 * HIDV * HIDV, bf2 + (size_t)fi0 * HIDV,
        Wf2 + (size_t)fi1 * HIDV * HIDV, bf2 + (size_t)fi1 * HIDV,
        Wf3 + (size_t)fi0 * (3 * KB - 1) * HIDV,
        bf3 + (size_t)fi0 * (3 * KB - 1),
        Wf3 + (size_t)fi1 * (3 * KB - 1) * HIDV,
        bf3 + (size_t)fi1 * (3 * KB - 1), hbuf, lane, c0, c1);

    // spline flow 0 (stage out tile through hbuf)
    lds_fence();
#pragma unroll
    for (int t = 0; t < 2; ++t)
#pragma unroll
      for (int v = 0; v < 8; ++v)
        hbuf[(t * 16 + v + mrow) * HSTR + r16] = c0[t][v];
    lds_fence();
    {
      float o[14];
#pragma unroll
      for (int c = 0; c < 14; ++c) o[c] = hbuf[lane * HSTR + c];
      x = rqs_step(x, o, &ld_sum);
    }

    // spline flow 1
    lds_fence();
#pragma unroll
    for (int t = 0; t < 2; ++t)
#pragma unroll
      for (int v = 0; v < 8; ++v)
        hbuf[(t * 16 + v + mrow) * HSTR + r16] = c1[t][v];
    lds_fence();
    {
      float o[14];
#pragma unroll
      for (int c = 0; c < 14; ++c) o[c] = hbuf[lane * HSTR + c];
      x = rqs_step(x, o, &ld_sum);
    }

    // ---- gaussianizer (mu, log sigma) ----
    {
      v8f cg[2];
      eval_gnet(aM, Wi1 + (size_t)jd * HIDV * CMAXV, bi1 + (size_t)jd * HIDV,
                Wi2 + (size_t)jd * HIDV * HIDV, bi2 + (size_t)jd * HIDV,
                Wi3 + (size_t)jd * 2 * HIDV, bi3 + (size_t)jd * 2, hbuf, lane,
                cg);
      lds_fence();
      if (r16 < 2) {
        float* dst = (r16 == 0) ? s_mu[wv] : s_ls[wv];
#pragma unroll
        for (int t = 0; t < 2; ++t)
#pragma unroll
          for (int v = 0; v < 8; ++v) dst[t * 16 + v + mrow] = cg[t][v];
      }
      lds_fence();

      float mu = s_mu[wv][lane];
      float logsig = s_ls[wv][lane];
      float sig = __expf(logsig);
      float z = (x - mu) / sig;
      float lp = -(z + __expf(-z)) - logsig;
      if (isnan(lp) || isinf(lp)) lp = -100.0f;
      total += ld_sum + lp;
    }
  }

  if (sample < N) out[sample] = total;
}

// ---------------------------------------------------------------------------
extern "C" void kernel_launch(void* const* d_in, const int* in_sizes, int n_in,
                              void* d_out, int out_size, void* d_ws,
                              size_t ws_size, hipStream_t stream) {
  (void)n_in;
  (void)d_ws;
  (void)ws_size;
  (void)out_size;
  const float* x_inp = (const float*)d_in[0];
  const float* cond_inp = (const float*)d_in[1];
  const float* Wi1 = (const float*)d_in[2];
  const float* bi1 = (const float*)d_in[3];
  const float* Wi2 = (const float*)d_in[4];
  const float* bi2 = (const float*)d_in[5];
  const float* Wi3 = (const float*)d_in[6];
  const float* bi3 = (const float*)d_in[7];
  const float* Wf1 = (const float*)d_in[8];
  const float* bf1 = (const float*)d_in[9];
  const float* Wf2 = (const float*)d_in[10];
  const float* bf2 = (const float*)d_in[11];
  const float* Wf3 = (const float*)d_in[12];
  const float* bf3 = (const float*)d_in[13];
  float* out = (float*)d_out;

  const int N = in_sizes[0] / DIMV;
  const int blocks = (N + 255) / 256;
  nsf_logp_kernel<<<blocks, 256, 0, stream>>>(x_inp, cond_inp, Wi1, bi1, Wi2,
                                              bi2, Wi3, bi3, Wf1, bf1, Wf2,
                                              bf2, Wf3, bf3, out, N);
}